// PromptEncoder_31456340475950
// MI455X (gfx1250) — compile-verified
//
#include <hip/hip_runtime.h>

typedef __attribute__((ext_vector_type(2))) float v2f;
typedef __attribute__((ext_vector_type(8))) float v8f;

// c = 2*((p+0.5)/1024) - 1  ==  (p+0.5)/512 - 1
#define INV512 (1.0f / 512.0f)

// NOTE on sin/cos: hardware v_sin_f32(u) computes sin(2*pi*u). The reference
// needs sin(2*pi*(c@gauss)), so we feed the *unscaled* c@gauss straight into
// v_sin_f32 / v_cos_f32 via the amdgcn builtins -- no 2*pi multiply, and no
// 1/(2*pi) pre-scale that __sinf would emit.

__global__ __launch_bounds__(256)
void prompt_encoder_kernel(const float* __restrict__ points,   // [B,128,2]
                           const int*   __restrict__ labels,   // [B,128]
                           const float* __restrict__ boxes,    // [B,4]
                           const float* __restrict__ gauss,    // [2,128]
                           const float* __restrict__ pe0,      // [256]
                           const float* __restrict__ pe1,      // [256]
                           const float* __restrict__ pe2,      // [256]
                           const float* __restrict__ pe3,      // [256]
                           const float* __restrict__ nap,      // [256]
                           float* __restrict__ out)            // [B,256]
{
    __shared__ __align__(16) float s_c[256];   // interleaved (cx,cy) per point
    __shared__ __align__(16) float s_g[256];   // interleaved (g0,g1) per feature
    __shared__ float s_sum[256];               // [0:128) sin sums, [128:256) cos sums
    __shared__ int   s_cnt[3];                 // label counts: -1,0,1 -> idx 0,1,2

    const int b = blockIdx.x;
    const int t = threadIdx.x;

    // ---- zero accumulators ----
    s_sum[t] = 0.0f;
    if (t < 3) s_cnt[t] = 0;
    __syncthreads();

    // ---- stage inputs into LDS ----
    if (t < 128) {
        const v2f p = *(const v2f*)(points + (size_t)(b * 128 + t) * 2);
        v2f c;
        c.x = (p.x + 0.5f) * INV512 - 1.0f;
        c.y = (p.y + 0.5f) * INV512 - 1.0f;
        *(v2f*)&s_c[2 * t] = c;                 // ds_store_b64
        const int lab = labels[b * 128 + t];    // in {-1,0,1}
        atomicAdd(&s_cnt[lab + 1], 1);
    } else {
        const int d = t - 128;
        v2f g;
        g.x = gauss[d];          // row 0
        g.y = gauss[128 + d];    // row 1
        *(v2f*)&s_g[2 * d] = g;  // ds_store_b64
    }
    __syncthreads();

    // ---- WMMA phase: u[m,f] = cx[m]*g0[f] + cy[m]*g1[f]  (K=2 padded to 4) ----
    const int  lane = t & 31;
    const int  w    = t >> 5;      // 8 waves; wave w owns points 16w..16w+15
    const bool lo   = lane < 16;
    const int  fl   = lane & 15;

    // A (16x4 f32): lanes 0-15 carry K=0,1 = (cx,cy); lanes 16-31 are K=2,3
    // and are zeroed ONCE here (v_cndmask, no exec branch). Because A's K=2,3
    // are zero, B's K=2,3 rows (upper lanes) may hold anything -> B needs no
    // masking at all.
    v2f A = *(const v2f*)&s_c[2 * (w * 16 + fl)];   // ds_load_b64
    A.x = lo ? A.x : 0.0f;
    A.y = lo ? A.y : 0.0f;

    // Preload ALL eight B tiles (16 VGPRs) so the inner loop has zero LDS
    // loads / dscnt waits: WMMA j+1 can issue while tile j's sin/cos TRANS
    // stream is still draining (XDL <-> VALU co-execution).
    v2f Bt[8];
    #pragma unroll
    for (int j = 0; j < 8; ++j)
        Bt[j] = *(const v2f*)&s_g[2 * (j * 16 + fl)];

    #pragma unroll
    for (int j = 0; j < 8; ++j) {
        const int f = j * 16 + fl;                  // feature column of this lane

        v8f acc = {};
        acc = __builtin_amdgcn_wmma_f32_16x16x4_f32(
            /*neg_a=*/false, A, /*neg_b=*/false, Bt[j],
            /*c_mod=*/(short)0, acc, /*reuse_a=*/false, /*reuse_b=*/false);

        // D tile: VGPR r = point row M=r (lanes 0-15) / M=r+8 (lanes 16-31).
        float ss = __builtin_amdgcn_sinf(acc[0]);
        float cs = __builtin_amdgcn_cosf(acc[0]);
        #pragma unroll
        for (int r = 1; r < 8; ++r) {
            ss += __builtin_amdgcn_sinf(acc[r]);
            cs += __builtin_amdgcn_cosf(acc[r]);
        }
        // both lane-halves accumulate their 8-point partials; ds_add_f32
        // resolves the 2-way conflict -- cheaper than permute + exec branch.
        atomicAdd(&s_sum[f],       ss);
        atomicAdd(&s_sum[f + 128], cs);
    }
    __syncthreads();

    // ---- box embedding + label bias + final mean over N+1 = 129 ----
    if (t < 128) {
        const float x0 = (boxes[b * 4 + 0] + 0.5f) * INV512 - 1.0f;
        const float y0 = (boxes[b * 4 + 1] + 0.5f) * INV512 - 1.0f;
        const float x1 = (boxes[b * 4 + 2] + 0.5f) * INV512 - 1.0f;
        const float y1 = (boxes[b * 4 + 3] + 0.5f) * INV512 - 1.0f;
        const v2f g = *(const v2f*)&s_g[2 * t];
        const float u0 = x0 * g.x + y0 * g.y;   // corner 0 (gets pe2)
        const float u1 = x1 * g.x + y1 * g.y;   // corner 1 (gets pe3)

        const float box_sin = 0.5f * ((__builtin_amdgcn_sinf(u0) + pe2[t]) +
                                      (__builtin_amdgcn_sinf(u1) + pe3[t]));
        const float box_cos = 0.5f * ((__builtin_amdgcn_cosf(u0) + pe2[t + 128]) +
                                      (__builtin_amdgcn_cosf(u1) + pe3[t + 128]));

        const float n_neg  = (float)s_cnt[0];   // label -1 -> nap
        const float n_zero = (float)s_cnt[1];   // label  0 -> pe1
        const float n_pos  = (float)s_cnt[2];   // label  1 -> pe0

        const float dl_sin = n_neg * nap[t]       + n_pos * pe0[t]       + n_zero * pe1[t];
        const float dl_cos = n_neg * nap[t + 128] + n_pos * pe0[t + 128] + n_zero * pe1[t + 128];

        const float inv = 1.0f / 129.0f;
        out[b * 256 + t]       = (s_sum[t]       + dl_sin + box_sin) * inv;
        out[b * 256 + t + 128] = (s_sum[t + 128] + dl_cos + box_cos) * inv;
    }
}

extern "C" void kernel_launch(void* const* d_in, const int* in_sizes, int n_in,
                              void* d_out, int out_size, void* d_ws, size_t ws_size,
                              hipStream_t stream) {
    const float* points = (const float*)d_in[0];
    const int*   labels = (const int*)  d_in[1];
    const float* boxes  = (const float*)d_in[2];
    const float* gauss  = (const float*)d_in[3];
    const float* pe0    = (const float*)d_in[4];
    const float* pe1    = (const float*)d_in[5];
    const float* pe2    = (const float*)d_in[6];
    const float* pe3    = (const float*)d_in[7];
    const float* nap    = (const float*)d_in[8];
    float* out          = (float*)d_out;

    const int B = in_sizes[1] / 128;  // labels is [B,128]

    prompt_encoder_kernel<<<B, 256, 0, stream>>>(points, labels, boxes, gauss,
                                                 pe0, pe1, pe2, pe3, nap, out);
}